// Radon_59691455480522
// MI455X (gfx1250) — compile-verified
//
#include <hip/hip_runtime.h>
#include <math.h>

// Radon forward transform, MI455X (gfx1250) tuned.
//
// Shapes: imgs (4,256,256) f32, angles (180,) f32, rays (256,4) f32
// out (4,180,256) f32.
//
// Strategy: gather-bound workload, not matrix math -> no WMMA. The CDNA5
// feature that matters is the 320KB-per-WGP LDS + async global->LDS DMA:
// each workgroup owns one (batch, angle) pair, async-copies the whole
// 256KB image into LDS (ASYNCcnt / s_wait_asynccnt), then does all 256x256
// bilinear samples out of LDS (ds_load), branch-free via clamp+select.

#define RES    256
#define NANG   180
#define NBATCH 4

typedef int v4i __attribute__((ext_vector_type(4)));
typedef __attribute__((address_space(1))) v4i g_v4i;   // global (AS1) 16B chunk
typedef __attribute__((address_space(3))) v4i l_v4i;   // LDS    (AS3) 16B chunk
typedef __attribute__((address_space(3))) char as3_char;

__device__ __forceinline__ void async_copy_b128(const void* gsrc, void* ldst) {
#if __has_builtin(__builtin_amdgcn_global_load_async_to_lds_b128)
    __builtin_amdgcn_global_load_async_to_lds_b128(
        (g_v4i*)gsrc, (l_v4i*)ldst, /*offset=*/0, /*cpol=*/0);
#else
    unsigned lds_off = (unsigned)(size_t)(as3_char*)ldst;
    asm volatile("global_load_async_to_lds_b128 %0, %1, off"
                 :: "v"(lds_off), "v"(gsrc) : "memory");
#endif
}

__device__ __forceinline__ void wait_async_zero() {
#if __has_builtin(__builtin_amdgcn_s_wait_asynccnt)
    __builtin_amdgcn_s_wait_asynccnt(0);
#else
    asm volatile("s_wait_asynccnt 0x0" ::: "memory");
#endif
}

__global__ void __launch_bounds__(RES)
radon_fwd_kernel(const float* __restrict__ imgs,
                 const float* __restrict__ angles,
                 const float* __restrict__ rays,
                 float* __restrict__ out)
{
    extern __shared__ float simg[];          // RES*RES floats = 256 KB LDS

    const int tid = threadIdx.x;             // ray index (0..255)
    const int blk = blockIdx.x;              // (batch, angle) pair
    const int b   = blk / NANG;
    const int a   = blk - b * NANG;

    // ---- Stage full image into LDS via async DMA (16B per op) ----------
    {
        const char* gbase = (const char*)(imgs + (size_t)b * RES * RES);
        char*       lbase = (char*)simg;
        const int   chunk0 = tid * 16;               // this lane's first 16B
        // RES*RES*4 bytes / (16B * 256 lanes) = 64 chunks per lane
        for (int j = 0; j < 64; ++j) {
            const int off = chunk0 + j * (256 * 16);
            async_copy_b128(gbase + off, lbase + off);
        }
    }
    wait_async_zero();       // drain this wave's ASYNCcnt
    __syncthreads();         // all 8 waves' DMA landed in LDS

    // ---- Per-(angle, ray) geometry -------------------------------------
    const float ang = angles[a];
    const float ca  = cosf(ang);
    const float sa  = sinf(ang);

    const int   r  = tid;
    const float sx = rays[r * 4 + 0];
    const float sy = rays[r * 4 + 1];
    const float ex = rays[r * 4 + 2];
    const float ey = rays[r * 4 + 3];

    const float rsx = sx * ca - sy * sa;
    const float rsy = sx * sa + sy * ca;
    const float rex = ex * ca - ey * sa;
    const float rey = ex * sa + ey * ca;

    const float inv_n = 1.0f / (float)RES;
    const float dfx = (rex - rsx) * inv_n;
    const float dfy = (rey - rsy) * inv_n;
    const float fx0 = rsx + 0.5f * dfx + ((float)(RES / 2) - 0.5f);
    const float fy0 = rsy + 0.5f * dfy + ((float)(RES / 2) - 0.5f);

    // ---- Integrate along the ray: 256 bilinear samples from LDS --------
    float acc = 0.0f;
#pragma unroll 4
    for (int k = 0; k < RES; ++k) {
        const float kf = (float)k;
        const float fx = fmaf(kf, dfx, fx0);
        const float fy = fmaf(kf, dfy, fy0);

        const float xf = floorf(fx);
        const float yf = floorf(fy);
        const int   x0 = (int)xf;
        const int   y0 = (int)yf;
        const float wx = fx - xf;
        const float wy = fy - yf;

        const int x1 = x0 + 1, y1 = y0 + 1;
        // clamped indices (loads always in-range -> no divergence)
        const int xc0 = x0 < 0 ? 0 : (x0 > RES - 1 ? RES - 1 : x0);
        const int xc1 = x1 < 0 ? 0 : (x1 > RES - 1 ? RES - 1 : x1);
        const int yc0 = y0 < 0 ? 0 : (y0 > RES - 1 ? RES - 1 : y0);
        const int yc1 = y1 < 0 ? 0 : (y1 > RES - 1 ? RES - 1 : y1);

        const bool vx0 = (unsigned)x0 < (unsigned)RES;
        const bool vx1 = (unsigned)x1 < (unsigned)RES;
        const bool vy0 = (unsigned)y0 < (unsigned)RES;
        const bool vy1 = (unsigned)y1 < (unsigned)RES;

        float v00 = simg[(yc0 << 8) + xc0];
        float v01 = simg[(yc0 << 8) + xc1];
        float v10 = simg[(yc1 << 8) + xc0];
        float v11 = simg[(yc1 << 8) + xc1];
        v00 = (vx0 & vy0) ? v00 : 0.0f;
        v01 = (vx1 & vy0) ? v01 : 0.0f;
        v10 = (vx0 & vy1) ? v10 : 0.0f;
        v11 = (vx1 & vy1) ? v11 : 0.0f;

        const float top = fmaf(wx, v01 - v00, v00);   // (1-wx)v00 + wx v01
        const float bot = fmaf(wx, v11 - v10, v10);
        acc = fmaf(wy, bot - top, acc + top);          // += (1-wy)top + wy bot
    }

    const float step = (ey - sy) * inv_n;              // L / n_steps
    out[((size_t)b * NANG + a) * RES + r] = acc * step;
}

extern "C" void kernel_launch(void* const* d_in, const int* in_sizes, int n_in,
                              void* d_out, int out_size, void* d_ws, size_t ws_size,
                              hipStream_t stream) {
    const float* imgs   = (const float*)d_in[0];
    const float* angles = (const float*)d_in[1];
    const float* rays   = (const float*)d_in[2];
    float*       out    = (float*)d_out;

    const dim3 grid(NBATCH * NANG);   // 720 workgroups, one (batch, angle) each
    const dim3 block(RES);            // 256 threads = 8 wave32, thread = ray
    const size_t shmem = (size_t)RES * RES * sizeof(float);  // 256 KB < 320 KB/WGP

    radon_fwd_kernel<<<grid, block, shmem, stream>>>(imgs, angles, rays, out);
}